// GAT_6854767804552
// MI455X (gfx1250) — compile-verified
//
#include <hip/hip_runtime.h>
#include <math.h>

#define NN 50000              // multiple of 16 (16*3125) -> full WMMA tiles
#define NE 1600000
#define EL (NE + NN)          // edges + self loops
#define NG 2048               // multiple of 16 -> full WMMA tiles
#define NEG_SLOPE 0.2f

typedef __attribute__((ext_vector_type(2))) float v2f;
typedef __attribute__((ext_vector_type(8))) float v8f;

// ---------------------------------------------------------------- utilities
__global__ void fill_kernel(float* __restrict__ p, float v, int n) {
  int i = blockIdx.x * blockDim.x + threadIdx.x;
  if (i < n) p[i] = v;
}

// pad x[NN,7] -> xp[NN,8] (col 7 = 0)
__global__ void pad_x_kernel(const float* __restrict__ x, float* __restrict__ xp) {
  int i = blockIdx.x * blockDim.x + threadIdx.x;
  if (i >= NN * 8) return;
  int n = i >> 3, c = i & 7;
  xp[i] = (c < 7) ? x[n * 7 + c] : 0.f;
}

__global__ void build_edges_kernel(const int* __restrict__ ei,
                                   int* __restrict__ srcE, int* __restrict__ dstE) {
  int i = blockIdx.x * blockDim.x + threadIdx.x;
  if (i < NE)      { srcE[i] = ei[i]; dstE[i] = ei[NE + i]; }
  else if (i < EL) { int n = i - NE; srcE[i] = n; dstE[i] = n; }
}

// ------------------------------------------------------------ WMMA f32 GEMM
// H[N,OUT] = X[N,K] @ W[K,OUT] (+ bias), X row-major with stride K.
// Requires N % 16 == 0 (true here: 50000, 2048) -> no store guards.
// One wave = one 16-col strip; B fragments register-cached; TILES row
// tiles of 16 per block.  blockDim = (OUT/16)*32 ; grid = ceil(N/(16*TILES)).
template<int K, int OUT, int TILES>
__global__ void gemm_wmma_kernel(const float* __restrict__ X,
                                 const float* __restrict__ W,
                                 const float* __restrict__ bias,
                                 float* __restrict__ H, int N) {
  const int lane    = threadIdx.x & 31;
  const int colBase = (threadIdx.x >> 5) * 16;
  const int mn   = lane & 15;          // A-row / B-col within tile
  const int koff = (lane >> 4) * 2;    // K sub-offset: 0 or 2
  const int col  = colBase + mn;

  // preload this wave's B strip into registers (reused across all row tiles)
  v2f bf[K / 4];
  #pragma unroll
  for (int kk = 0; kk < K / 4; ++kk) {
    const int ka = kk * 4 + koff;
    bf[kk].x = W[(long)ka * OUT + col];
    bf[kk].y = W[(long)(ka + 1) * OUT + col];
  }
  const float bv = bias ? bias[col] : 0.f;
  const int half = (lane >> 4) * 8;    // C/D layout: lanes16-31 hold M=8..15

  for (int t = 0; t < TILES; ++t) {
    const int rowBase = (blockIdx.x * TILES + t) * 16;
    if (rowBase >= N) break;                  // wave-uniform exit only
    const float* xr = X + (long)(rowBase + mn) * K;
    v8f acc = {};
    #pragma unroll
    for (int kk = 0; kk < K / 4; ++kk) {
      v2f a = *(const v2f*)(xr + kk * 4 + koff);   // one b64 load
      acc = __builtin_amdgcn_wmma_f32_16x16x4_f32(false, a, false, bf[kk],
                                                  (short)0, acc, false, false);
    }
    float* hout = H + (long)(rowBase + half) * OUT + col;
    #pragma unroll
    for (int r = 0; r < 8; ++r)                // full tile: unguarded stores
      hout[(long)r * OUT] = acc[r] + bv;
  }
}

// ------------------------------------------------- per-node attention logits
// als[n] = h[n,:].a_src ; ald[n] = h[n,:].a_dst  (one wave32 per node)
__global__ void attn_kernel(const float* __restrict__ h,
                            const float* __restrict__ asv,
                            const float* __restrict__ adv,
                            float* __restrict__ als, float* __restrict__ ald,
                            int OUT) {
  int node = blockIdx.x * 8 + (threadIdx.x >> 5);
  int lane = threadIdx.x & 31;
  if (node >= NN) return;
  float ss = 0.f, sd = 0.f;
  for (int d = lane; d < OUT; d += 32) {
    float hv = h[(long)node * OUT + d];
    ss += hv * asv[d];
    sd += hv * adv[d];
  }
  for (int o = 16; o; o >>= 1) {
    ss += __shfl_down(ss, o, 32);
    sd += __shfl_down(sd, o, 32);
  }
  if (lane == 0) { als[node] = ss; ald[node] = sd; }
}

// ---------------------------------------------------------------- edge passes
__device__ inline void atomicMaxF(float* a, float v) {
  if (v >= 0.f) atomicMax((int*)a, __float_as_int(v));
  else          atomicMin((unsigned int*)a, (unsigned int)__float_as_int(v));
}

__global__ void edge_max_kernel(const int* __restrict__ srcE, const int* __restrict__ dstE,
                                const float* __restrict__ als, const float* __restrict__ ald,
                                float* __restrict__ e, float* __restrict__ m) {
  int i = blockIdx.x * blockDim.x + threadIdx.x;
  if (i >= EL) return;
  float v = als[srcE[i]] + ald[dstE[i]];
  v = (v > 0.f) ? v : v * NEG_SLOPE;      // leaky_relu
  e[i] = v;
  atomicMaxF(&m[dstE[i]], v);
}

__global__ void edge_expsum_kernel(const int* __restrict__ dstE,
                                   const float* __restrict__ m,
                                   float* __restrict__ e, float* __restrict__ den) {
  int i = blockIdx.x * blockDim.x + threadIdx.x;
  if (i >= EL) return;
  int t = dstE[i];
  float ex = expf(e[i] - m[t]);
  e[i] = ex;
  atomicAdd(&den[t], ex);
}

// out[dst] += alpha * h[src] ; OUT/4 threads per edge, b128 gathers.
template<int OUT>
__global__ void edge_scatter_kernel(const int* __restrict__ srcE, const int* __restrict__ dstE,
                                    const float* __restrict__ e, const float* __restrict__ den,
                                    const float* __restrict__ h, float* __restrict__ out) {
  const int TPE = OUT / 4;                 // threads per edge
  const int epb = 256 / TPE;               // edges per block
  int edge = blockIdx.x * epb + (int)(threadIdx.x / TPE);
  int d4   = (threadIdx.x % TPE) * 4;
  if (edge >= EL) return;
  int s = srcE[edge], t = dstE[edge];
  float alpha = e[edge] / den[t];
  float4 hv = *(const float4*)(h + (long)s * OUT + d4);   // one b128 load
  float* o = out + (long)t * OUT + d4;
  atomicAdd(o + 0, alpha * hv.x);
  atomicAdd(o + 1, alpha * hv.y);
  atomicAdd(o + 2, alpha * hv.z);
  atomicAdd(o + 3, alpha * hv.w);
}

__global__ void elu_bias_kernel(float* __restrict__ x, const float* __restrict__ bias,
                                int total, int OUT) {
  int i = blockIdx.x * blockDim.x + threadIdx.x;
  if (i >= total) return;
  float v = x[i] + bias[i & (OUT - 1)];   // OUT is a power of two (128/64)
  x[i] = (v > 0.f) ? v : expm1f(v);       // elu(alpha=1)
}

// ------------------------------------------------------------------- pooling
__global__ void pool_kernel(const float* __restrict__ x, const int* __restrict__ batch,
                            float* __restrict__ gsum, float* __restrict__ gcnt) {
  int i = blockIdx.x * blockDim.x + threadIdx.x;
  if (i >= NN * 64) return;
  int node = i >> 6, d = i & 63;
  int g = batch[node];
  atomicAdd(&gsum[g * 64 + d], x[i]);
  if (d == 0) atomicAdd(&gcnt[g], 1.f);
}

__global__ void finalize_kernel(const float* __restrict__ gsum, const float* __restrict__ gcnt,
                                const float* __restrict__ xn, float* __restrict__ gfeat) {
  int i = blockIdx.x * blockDim.x + threadIdx.x;
  if (i >= NG * 80) return;
  int g = i / 80, j = i % 80;
  gfeat[i] = (j < 64) ? gsum[g * 64 + j] / fmaxf(gcnt[g], 1.f)
                      : xn[g * 16 + (j - 64)];
}

// ------------------------------------------------------------------- driver
struct Bufs {
  float *A, *B, *als, *ald, *e, *mv, *den;
  int   *srcE, *dstE;
  float *gsum, *gcnt, *gfeat;
};

// softmax-attention message passing + ELU for one layer (h already in B)
template<int OUT>
static void run_edges(const Bufs& bb, const float* as, const float* ad,
                      const float* bias, hipStream_t stream) {
  const int T = 256;
  attn_kernel<<<(NN + 7) / 8, 256, 0, stream>>>(bb.B, as, ad, bb.als, bb.ald, OUT);
  fill_kernel<<<(NN + T - 1) / T, T, 0, stream>>>(bb.mv, -INFINITY, NN);
  fill_kernel<<<(NN + T - 1) / T, T, 0, stream>>>(bb.den, 0.f, NN);
  fill_kernel<<<(NN * OUT + T - 1) / T, T, 0, stream>>>(bb.A, 0.f, NN * OUT);
  edge_max_kernel<<<(EL + T - 1) / T, T, 0, stream>>>(bb.srcE, bb.dstE, bb.als, bb.ald, bb.e, bb.mv);
  edge_expsum_kernel<<<(EL + T - 1) / T, T, 0, stream>>>(bb.dstE, bb.mv, bb.e, bb.den);
  const int epb = 256 / (OUT / 4);
  edge_scatter_kernel<OUT><<<(EL + epb - 1) / epb, 256, 0, stream>>>(
      bb.srcE, bb.dstE, bb.e, bb.den, bb.B, bb.A);
  elu_bias_kernel<<<(NN * OUT + T - 1) / T, T, 0, stream>>>(bb.A, bias, NN * OUT, OUT);
}

static void run_branch(const float* x_in, const int* edge_index,
                       const int* batch, const float* xnorm2,
                       void* const* d_in, const float* linW, const float* linb,
                       const Bufs& bb, float* out, hipStream_t stream) {
  const int T = 256;
  build_edges_kernel<<<(EL + T - 1) / T, T, 0, stream>>>(edge_index, bb.srcE, bb.dstE);

  // ---- layer 1: pad x to K=8, GEMM 8->128
  pad_x_kernel<<<(NN * 8 + T - 1) / T, T, 0, stream>>>(x_in, bb.A);
  gemm_wmma_kernel<8, 128, 8><<<(NN + 127) / 128, 256, 0, stream>>>(
      bb.A, (const float*)d_in[8], nullptr, bb.B, NN);
  run_edges<128>(bb, (const float*)d_in[9], (const float*)d_in[10],
                 (const float*)d_in[11], stream);

  // ---- layer 2: 128 -> 128
  gemm_wmma_kernel<128, 128, 8><<<(NN + 127) / 128, 256, 0, stream>>>(
      bb.A, (const float*)d_in[12], nullptr, bb.B, NN);
  run_edges<128>(bb, (const float*)d_in[13], (const float*)d_in[14],
                 (const float*)d_in[15], stream);

  // ---- layer 3: 128 -> 64
  gemm_wmma_kernel<128, 64, 8><<<(NN + 127) / 128, 128, 0, stream>>>(
      bb.A, (const float*)d_in[16], nullptr, bb.B, NN);
  run_edges<64>(bb, (const float*)d_in[17], (const float*)d_in[18],
                (const float*)d_in[19], stream);

  // ---- global mean pool + concat + linear head
  fill_kernel<<<(NG * 64 + T - 1) / T, T, 0, stream>>>(bb.gsum, 0.f, NG * 64);
  fill_kernel<<<(NG + T - 1) / T, T, 0, stream>>>(bb.gcnt, 0.f, NG);
  pool_kernel<<<(NN * 64 + T - 1) / T, T, 0, stream>>>(bb.A, batch, bb.gsum, bb.gcnt);
  finalize_kernel<<<(NG * 80 + T - 1) / T, T, 0, stream>>>(bb.gsum, bb.gcnt, xnorm2, bb.gfeat);
  gemm_wmma_kernel<80, 64, 8><<<(NG + 127) / 128, 128, 0, stream>>>(
      bb.gfeat, linW, linb, out, NG);
}

extern "C" void kernel_launch(void* const* d_in, const int* in_sizes, int n_in,
                              void* d_out, int out_size, void* d_ws, size_t ws_size,
                              hipStream_t stream) {
  const float* x1    = (const float*)d_in[0];
  const float* x2    = (const float*)d_in[1];
  const int*   ei1   = (const int*)d_in[2];
  const int*   ei2   = (const int*)d_in[3];
  const int*   batch = (const int*)d_in[4];
  // d_in[5] = half_y (unused by the reference computation)
  const float* xn1   = (const float*)d_in[6];
  const float* xn2   = (const float*)d_in[7];
  const float* linW  = (const float*)d_in[20];
  const float* linb  = (const float*)d_in[21];

  // workspace carve-up
  char* w = (char*)d_ws;
  Bufs bb;
  bb.A     = (float*)w; w += (size_t)NN * 128 * 4;
  bb.B     = (float*)w; w += (size_t)NN * 128 * 4;
  bb.als   = (float*)w; w += (size_t)NN * 4;
  bb.ald   = (float*)w; w += (size_t)NN * 4;
  bb.e     = (float*)w; w += (size_t)EL * 4;
  bb.mv    = (float*)w; w += (size_t)NN * 4;
  bb.den   = (float*)w; w += (size_t)NN * 4;
  bb.srcE  = (int*)w;   w += (size_t)EL * 4;
  bb.dstE  = (int*)w;   w += (size_t)EL * 4;
  bb.gsum  = (float*)w; w += (size_t)NG * 64 * 4;
  bb.gcnt  = (float*)w; w += (size_t)NG * 4;
  bb.gfeat = (float*)w; w += (size_t)NG * 80 * 4;

  float* out = (float*)d_out;
  run_branch(x1, ei1, batch, xn1, d_in, linW, linb, bb, out, stream);
  run_branch(x2, ei2, batch, xn2, d_in, linW, linb, bb, out + (size_t)NG * 64, stream);
}